// CTRNN_54494545052030
// MI455X (gfx1250) — compile-verified
//
#include <hip/hip_runtime.h>

typedef float v2f __attribute__((ext_vector_type(2)));
typedef float v8f __attribute__((ext_vector_type(8)));

#define CTRNN_ALPHA 0.1f

// ---------------------------------------------------------------------------
// Phase 1: xproj[m, n] = sum_k x[m,k] * Wx[n,k] + bx[n]
// One wave (32 threads) computes one 16x16 tile with V_WMMA_F32_16X16X4_F32.
// A-frag: lane (half,lid) holds A[m0+lid, k+2*half .. +1]  (contiguous float2)
// B-frag: lane (half,lid) holds B[k+2*half.., n0+lid] = Wx[n0+lid, k+2*half..]
// ---------------------------------------------------------------------------
__global__ __launch_bounds__(32) void ctrnn_xproj_wmma(
    const float* __restrict__ x,   // [M, K] row-major (M = S*B, K = I)
    const float* __restrict__ Wx,  // [N, K] row-major (N = H)
    const float* __restrict__ bx,  // [N]
    float* __restrict__ xproj,     // [M, N]
    int M, int K, int N)
{
  const int lane = threadIdx.x & 31;
  const int half = lane >> 4;
  const int lid  = lane & 15;
  const int n0 = blockIdx.x << 4;
  const int m0 = blockIdx.y << 4;

  const float* __restrict__ arow = x  + (size_t)(m0 + lid) * K + 2 * half;
  const float* __restrict__ brow = Wx + (size_t)(n0 + lid) * K + 2 * half;

  v8f c = {};
#pragma unroll 8
  for (int k = 0; k < K; k += 4) {
    v2f a = *(const v2f*)(arow + k);
    v2f b = *(const v2f*)(brow + k);
    // 8 args: (neg_a, A, neg_b, B, c_mod, C, reuse_a, reuse_b)
    c = __builtin_amdgcn_wmma_f32_16x16x4_f32(false, a, false, b,
                                              (short)0, c, false, false);
  }

  const float bias = bx[n0 + lid];  // per-column bias, same for all 8 rows
#pragma unroll
  for (int r = 0; r < 8; ++r) {
    const int m = m0 + r + 8 * half;
    xproj[(size_t)m * N + (n0 + lid)] = c[r] + bias;
  }
}

// ---------------------------------------------------------------------------
// Phase 2: one recurrent step (launched once per timestep).
//   pre  = xproj_t + h_prev @ Wh^T + bh
//   h_t  = (1-ALPHA)*h_prev + ALPHA*relu(pre)
// out_t holds xproj_t on entry and h_t on exit (in-place, read-before-write).
// ---------------------------------------------------------------------------
__global__ __launch_bounds__(32) void ctrnn_step_wmma(
    const float* __restrict__ h_prev,  // [64, 1024]
    const float* __restrict__ Wh,      // [1024, 1024] row-major
    const float* __restrict__ bh,      // [1024]
    float* __restrict__ out_t)         // [64, 1024]: in xproj_t, out h_t
{
  const int H = 1024;
  const int lane = threadIdx.x & 31;
  const int half = lane >> 4;
  const int lid  = lane & 15;
  const int n0 = blockIdx.x << 4;
  const int m0 = blockIdx.y << 4;

  const float* __restrict__ arow = h_prev + (size_t)(m0 + lid) * H + 2 * half;
  const float* __restrict__ brow = Wh     + (size_t)(n0 + lid) * H + 2 * half;

  v8f c = {};
#pragma unroll 8
  for (int k = 0; k < H; k += 4) {
    v2f a = *(const v2f*)(arow + k);
    v2f b = *(const v2f*)(brow + k);
    c = __builtin_amdgcn_wmma_f32_16x16x4_f32(false, a, false, b,
                                              (short)0, c, false, false);
  }

  const float bias = bh[n0 + lid];
#pragma unroll
  for (int r = 0; r < 8; ++r) {
    const int m = m0 + r + 8 * half;
    const size_t idx = (size_t)m * H + (n0 + lid);
    const float xp = out_t[idx];     // xproj_t (not yet overwritten)
    const float hp = h_prev[idx];
    const float hn = fmaxf(xp + c[r] + bias, 0.0f);
    out_t[idx] = hp * (1.0f - CTRNN_ALPHA) + hn * CTRNN_ALPHA;
  }
}

// Tail: copy h at t = S-1 into the h_final slot of d_out.
__global__ void ctrnn_copy_final(const float* __restrict__ src,
                                 float* __restrict__ dst, int n)
{
  const int i = blockIdx.x * blockDim.x + threadIdx.x;
  if (i < n) dst[i] = src[i];
}

extern "C" void kernel_launch(void* const* d_in, const int* in_sizes, int n_in,
                              void* d_out, int out_size, void* d_ws, size_t ws_size,
                              hipStream_t stream) {
  (void)in_sizes; (void)n_in; (void)out_size; (void)d_ws; (void)ws_size;

  const float* x  = (const float*)d_in[0];  // [512, 64, 256]
  const float* h0 = (const float*)d_in[1];  // [64, 1024] (zeros)
  const float* Wx = (const float*)d_in[2];  // [1024, 256]
  const float* bx = (const float*)d_in[3];  // [1024]
  const float* Wh = (const float*)d_in[4];  // [1024, 1024]
  const float* bh = (const float*)d_in[5];  // [1024]
  float* out = (float*)d_out;               // [512*64*1024] output + [64*1024] h_final

  const int S = 512, B = 64, I = 256, H = 1024;
  const int M = S * B;

  // Phase 1: xproj for all timesteps, written straight into d_out's output slots.
  dim3 g1(H / 16, M / 16);
  ctrnn_xproj_wmma<<<g1, dim3(32), 0, stream>>>(x, Wx, bx, out, M, I, H);

  // Phase 2: sequential scan; one kernel per step is the device-wide sync.
  dim3 g2(H / 16, B / 16);
  for (int t = 0; t < S; ++t) {
    const float* hprev = (t == 0) ? h0 : (out + (size_t)(t - 1) * B * H);
    ctrnn_step_wmma<<<g2, dim3(32), 0, stream>>>(hprev, Wh, bh,
                                                 out + (size_t)t * B * H);
  }

  // h_final = h[S-1]
  const int n = B * H;
  ctrnn_copy_final<<<(n + 255) / 256, 256, 0, stream>>>(
      out + (size_t)(S - 1) * B * H, out + (size_t)S * B * H, n);
}